// NodeAttention_53884659695800
// MI455X (gfx1250) — compile-verified
//
#include <hip/hip_runtime.h>
#include <stdint.h>

// NodeAttention on MI455X (gfx1250).
// Memory-bound streaming op (192 MiB @ 23.3 TB/s ~= 8.2 us floor).
// softmax over a length-1 axis == 1  =>  embed_e == x_in, so per row we only
// need the scalar gate c = sigmoid(nf.w_lo + x.w_hi + b) and out = (1-c)x + cx,
// written transposed [B,S,H] -> [S,B,H].
// CDNA5 path: TDM tensor_load_to_lds double-buffered tiles + s_wait_tensorcnt,
// wave32 shuffle reduction, non-temporal b128 stores.

typedef uint32_t u32;
typedef uint64_t u64;
typedef float v4f __attribute__((ext_vector_type(4)));
typedef u32   v4u __attribute__((ext_vector_type(4)));
typedef int   v4i __attribute__((ext_vector_type(4)));
typedef int   v8i __attribute__((ext_vector_type(8)));

#define B_     8
#define S_     4096
#define H_     512
#define N_     (B_ * S_)     // 32768 rows
#define R_     8             // rows per TDM tile (== waves per block)
#define NT_    (N_ / R_)     // 4096 tiles
#define GRID_  1024
#define BLOCK_ 256

// Build a 2-D Tensor DMA descriptor (D#) and issue TENSOR_LOAD_TO_LDS.
// Bitfields per CDNA5 ISA.
__device__ __forceinline__ void tdm_load_2d(const void* gptr, u32 lds_off,
                                            u32 w_elems, u32 rows,
                                            u32 stride_elems) {
  u64 ga = (u64)(uintptr_t)gptr;
  v4u g0;
  g0.x = 1u;                                              // count=1 (valid), no gather
  g0.y = lds_off;                                         // lds_addr [63:32]
  g0.z = (u32)ga;                                         // global_addr lo
  g0.w = (u32)((ga >> 32) & 0x01FFFFFFu) | (2u << 30);    // addr[56:32] | type=2
  v8i g1;
  g1[0] = (int)(2u << 16);                                     // data_size=2 (4B)
  g1[1] = (int)((w_elems & 0xFFFFu) << 16);                    // tensor_dim0[15:0]
  g1[2] = (int)((w_elems >> 16) | ((rows & 0xFFFFu) << 16));   // dim0[31:16]|dim1[15:0]
  g1[3] = (int)((rows >> 16) | ((w_elems & 0xFFFFu) << 16));   // dim1[31:16]|tile_dim0
  g1[4] = (int)(rows & 0xFFFFu);                               // tile_dim1, tile_dim2=0
  g1[5] = (int)stride_elems;                                   // tensor_dim0_stride lo32
  g1[6] = 0;
  g1[7] = 0;
  v4i zg = {0, 0, 0, 0};                                       // 2-D: groups 2/3 unused
#if __clang_major__ >= 23
  v8i zg8 = {0, 0, 0, 0, 0, 0, 0, 0};
  __builtin_amdgcn_tensor_load_to_lds(g0, g1, zg, zg, zg8, 0);
#else
  __builtin_amdgcn_tensor_load_to_lds(g0, g1, zg, zg, 0);
#endif
}

__device__ __forceinline__ u32 lds_off_of(const void* p) {
  // LDS flat address = {SHARED_BASE, offset[31:0]} -> low 32 bits are the offset.
  return (u32)(uintptr_t)p;
}

__global__ __launch_bounds__(BLOCK_) void node_attn_tdm(
    const float* __restrict__ x_in, const float* __restrict__ eoa,
    const float* __restrict__ dnd, const float* __restrict__ wih,
    const float* __restrict__ bih, float* __restrict__ out) {
  __shared__ float sx[2][R_][H_];        // 32 KB
  __shared__ float se[2][R_][H_ / 2];    // 16 KB
  __shared__ float sd[2][R_][H_ / 2];    // 16 KB

  const int lane = threadIdx.x & 31;
  const int wv = threadIdx.x >> 5;
  const bool w0 = (threadIdx.x < 32);

  // Per-lane weight slice: columns {j*128 + 4*lane .. +3}, j = 0..3.
  v4f wlo[4], whi[4];
#pragma unroll
  for (int j = 0; j < 4; ++j) {
    wlo[j] = *(const v4f*)(wih + j * 128 + lane * 4);
    whi[j] = *(const v4f*)(wih + H_ + j * 128 + lane * 4);
  }
  const float bias = bih[0];

  int tile = (int)blockIdx.x;
  if (w0) {  // prologue: tile -> buffer 0 (wave-uniform branch; TDM ignores EXEC)
    tdm_load_2d(x_in + (size_t)tile * R_ * H_, lds_off_of(&sx[0][0][0]), H_, R_, H_);
    tdm_load_2d(eoa + (size_t)tile * R_ * (H_ / 2), lds_off_of(&se[0][0][0]), H_ / 2, R_, H_ / 2);
    tdm_load_2d(dnd + (size_t)tile * R_ * (H_ / 2), lds_off_of(&sd[0][0][0]), H_ / 2, R_, H_ / 2);
  }

  int buf = 0;
  for (; tile < NT_; tile += GRID_) {
    const int nxt = tile + GRID_;
    if (w0) {
      if (nxt < NT_) {  // prefetch next tile into the other buffer, keep 3 in flight
        tdm_load_2d(x_in + (size_t)nxt * R_ * H_, lds_off_of(&sx[buf ^ 1][0][0]), H_, R_, H_);
        tdm_load_2d(eoa + (size_t)nxt * R_ * (H_ / 2), lds_off_of(&se[buf ^ 1][0][0]), H_ / 2, R_, H_ / 2);
        tdm_load_2d(dnd + (size_t)nxt * R_ * (H_ / 2), lds_off_of(&sd[buf ^ 1][0][0]), H_ / 2, R_, H_ / 2);
        __builtin_amdgcn_s_wait_tensorcnt(3);  // current tile's 3 DMAs retired (in-order)
      } else {
        __builtin_amdgcn_s_wait_tensorcnt(0);
      }
    }
    __syncthreads();  // tile data visible to all waves

    // wave wv owns row (tile*R_ + wv); lane covers 4 strided float4 chunks.
    v4f xr[4];
    float acc = 0.0f;
#pragma unroll
    for (int j = 0; j < 4; ++j) {
      xr[j] = *(const v4f*)&sx[buf][wv][j * 128 + lane * 4];
      v4f nf = (j < 2) ? *(const v4f*)&se[buf][wv][j * 128 + lane * 4]
                       : *(const v4f*)&sd[buf][wv][(j - 2) * 128 + lane * 4];
      acc = fmaf(nf.x, wlo[j].x, acc);
      acc = fmaf(nf.y, wlo[j].y, acc);
      acc = fmaf(nf.z, wlo[j].z, acc);
      acc = fmaf(nf.w, wlo[j].w, acc);
      acc = fmaf(xr[j].x, whi[j].x, acc);
      acc = fmaf(xr[j].y, whi[j].y, acc);
      acc = fmaf(xr[j].z, whi[j].z, acc);
      acc = fmaf(xr[j].w, whi[j].w, acc);
    }
    // wave32 butterfly reduction
#pragma unroll
    for (int m = 16; m >= 1; m >>= 1) acc += __shfl_xor(acc, m, 32);

    const float cg = 1.0f / (1.0f + expf(-(acc + bias)));
    const float om = 1.0f - cg;

    const int n = tile * R_ + wv;
    const int bb = n >> 12;       // n / S_
    const int ss = n & (S_ - 1);  // n % S_
    float* op = out + (((size_t)ss * B_ + bb) << 9);  // * H_
#pragma unroll
    for (int j = 0; j < 4; ++j) {
      v4f o;
      o.x = om * xr[j].x + cg * xr[j].x;
      o.y = om * xr[j].y + cg * xr[j].y;
      o.z = om * xr[j].z + cg * xr[j].z;
      o.w = om * xr[j].w + cg * xr[j].w;
      __builtin_nontemporal_store(o, (v4f*)(op + j * 128 + lane * 4));
    }
    __syncthreads();  // everyone done reading buf before it becomes the DMA target
    buf ^= 1;
  }
}

extern "C" void kernel_launch(void* const* d_in, const int* in_sizes, int n_in,
                              void* d_out, int out_size, void* d_ws,
                              size_t ws_size, hipStream_t stream) {
  (void)in_sizes; (void)n_in; (void)out_size; (void)d_ws; (void)ws_size;
  const float* x_in = (const float*)d_in[0];
  const float* eoa = (const float*)d_in[1];
  const float* dnd = (const float*)d_in[2];
  const float* wih = (const float*)d_in[3];
  const float* bih = (const float*)d_in[4];
  float* out = (float*)d_out;
  hipLaunchKernelGGL(node_attn_tdm, dim3(GRID_), dim3(BLOCK_), 0, stream,
                     x_in, eoa, dnd, wih, bih, out);
}